// WriteUnit_28535762714868
// MI455X (gfx1250) — compile-verified
//
#include <hip/hip_runtime.h>
#include <hip/hip_bf16.h>
#include <math.h>

typedef __attribute__((ext_vector_type(2))) float v2f;
typedef __attribute__((ext_vector_type(4))) float v4f;
typedef __attribute__((ext_vector_type(8))) float v8f;

#define DD    512
#define TT    12
#define BB    8192
#define BROWS 16     // rows of B per block
#define NTHR  256    // 8 wave32s

__device__ __forceinline__ v8f wmma_f32(v2f a, v2f b, v8f c) {
    // V_WMMA_F32_16X16X4_F32 : D = A(16x4) * B(4x16) + C(16x16), fp32 throughout
    return __builtin_amdgcn_wmma_f32_16x16x4_f32(
        /*neg_a=*/false, a, /*neg_b=*/false, b,
        /*c_mod=*/(short)0, c, /*reuse_a=*/false, /*reuse_b=*/false);
}

// ---------------------------------------------------------------------------
// Kernel 1: selfControl GEMM + attention logits + softmax + selfSmry
// ---------------------------------------------------------------------------
__global__ __launch_bounds__(NTHR) void wu_attn_kernel(
    const float* __restrict__ contControl,   // [B, D]
    const float* __restrict__ allControls,   // [B, T, D]
    const float* __restrict__ allMemories,   // [B, T, D]
    const float* __restrict__ Wc,            // [D, D] (out, in)
    const float* __restrict__ bc,            // [D]
    const float* __restrict__ Wa,            // [1, D]
    const float* __restrict__ ba,            // [1]
    float* __restrict__ selfSmry)            // [B, D] (workspace)
{
    __shared__ float sBuf[BROWS * DD];       // contControl tile, then u = selfControl*Wa
    __shared__ float sL[BROWS * TT];         // logits -> attention weights

    const int tid     = threadIdx.x;
    const int lane    = tid & 31;
    const int w       = tid >> 5;            // wave id 0..7
    const int L       = lane & 15;
    const int hi      = lane >> 4;
    const int rowBase = blockIdx.x * BROWS;

    // ---- Phase 1: stage contControl[rowBase:rowBase+16, :] into LDS --------
    {
        const v4f* src = (const v4f*)(contControl + (long)rowBase * DD);
        v4f* dst = (v4f*)sBuf;
        #pragma unroll
        for (int i = tid; i < BROWS * DD / 4; i += NTHR) dst[i] = src[i];
    }
    __syncthreads();

    // ---- Phase 2: selfControl = contControl @ Wc.T + bc  (wave: 16x64) ----
    const int nB = w * 64;
    v8f acc[4];
    #pragma unroll
    for (int j = 0; j < 4; ++j) {
        const float bias = bc[nB + j * 16 + L];
        #pragma unroll
        for (int r = 0; r < 8; ++r) acc[j][r] = bias;
    }
    #pragma unroll 4
    for (int k0 = 0; k0 < DD; k0 += 4) {
        const v2f a = *(const v2f*)&sBuf[L * DD + k0 + 2 * hi];
        #pragma unroll
        for (int j = 0; j < 4; ++j) {
            const int n = nB + j * 16 + L;
            const v2f b = *(const v2f*)&Wc[(long)n * DD + k0 + 2 * hi];
            acc[j] = wmma_f32(a, b, acc[j]);
        }
    }
    __syncthreads();   // everyone done reading contControl tile

    // ---- Phase 3: u[m,n] = selfControl[m,n] * Wa[n]  into LDS -------------
    #pragma unroll
    for (int j = 0; j < 4; ++j) {
        const int n = nB + j * 16 + L;
        const float wa = Wa[n];
        #pragma unroll
        for (int r = 0; r < 8; ++r) {
            const int m = 8 * hi + r;
            sBuf[m * DD + n] = acc[j][r] * wa;
        }
    }
    __syncthreads();

    // ---- Phase 4: logits[m,t] = sum_d u[m,d]*allControls[b,t,d] + ba ------
    // 192 (row,t) pairs, 24 per wave, lane-parallel over d + shfl reduction
    for (int i = 0; i < 24; ++i) {
        const int p = w * 24 + i;
        const int m = p / TT;
        const int t = p % TT;
        const float* ac = allControls + ((long)(rowBase + m) * TT + t) * DD;
        float s = 0.f;
        #pragma unroll
        for (int d0 = lane; d0 < DD; d0 += 32) s += sBuf[m * DD + d0] * ac[d0];
        #pragma unroll
        for (int off = 16; off > 0; off >>= 1) s += __shfl_xor(s, off, 32);
        if (lane == 0) sL[m * TT + t] = s + ba[0];
    }
    __syncthreads();

    // ---- Phase 5: softmax over T per row -----------------------------------
    if (tid < BROWS) {
        float l[TT];
        float mx = -1e30f;
        #pragma unroll
        for (int t = 0; t < TT; ++t) { l[t] = sL[tid * TT + t]; mx = fmaxf(mx, l[t]); }
        float sum = 0.f;
        #pragma unroll
        for (int t = 0; t < TT; ++t) { l[t] = expf(l[t] - mx); sum += l[t]; }
        const float inv = 1.f / sum;
        #pragma unroll
        for (int t = 0; t < TT; ++t) sL[tid * TT + t] = l[t] * inv;
    }
    __syncthreads();

    // ---- Phase 6: selfSmry[b,d] = sum_t attn[b,t] * allMemories[b,t,d] ----
    for (int idx = tid; idx < BROWS * DD; idx += NTHR) {
        const int m  = idx >> 9;          // / DD
        const int d0 = idx & (DD - 1);
        const float* am = allMemories + ((long)(rowBase + m) * TT) * DD + d0;
        float s = 0.f;
        #pragma unroll
        for (int t = 0; t < TT; ++t) s += sL[m * TT + t] * am[(long)t * DD];
        selfSmry[(long)(rowBase + m) * DD + d0] = s;
    }
}

// ---------------------------------------------------------------------------
// Kernel 2: newMemory = [memory|info|selfSmry] @ Wl.T + bl ;
//           z = sigmoid(control @ Wg.T + bg) ; out = nm*z + memory*(1-z)
// ---------------------------------------------------------------------------
__global__ __launch_bounds__(NTHR) void wu_write_kernel(
    const float* __restrict__ memory,     // [B, D]
    const float* __restrict__ info,       // [B, D]
    const float* __restrict__ control,    // [B, D]
    const float* __restrict__ selfSmry,   // [B, D]
    const float* __restrict__ Wl,         // [D, 3D]
    const float* __restrict__ bl,         // [D]
    const float* __restrict__ Wg,         // [D, D]
    const float* __restrict__ bg,         // [D]
    float* __restrict__ out)              // [B, D]
{
    __shared__ float sM[BROWS * DD];      // memory tile (persistent, reused for gate)
    __shared__ float sX[BROWS * DD];      // staging: info -> selfSmry -> control

    const int tid     = threadIdx.x;
    const int lane    = tid & 31;
    const int w       = tid >> 5;
    const int L       = lane & 15;
    const int hi      = lane >> 4;
    const int rowBase = blockIdx.x * BROWS;
    const int nB      = w * 64;

    // stage memory + info
    {
        const v4f* srcM = (const v4f*)(memory + (long)rowBase * DD);
        const v4f* srcI = (const v4f*)(info   + (long)rowBase * DD);
        v4f* dM = (v4f*)sM;
        v4f* dX = (v4f*)sX;
        #pragma unroll
        for (int i = tid; i < BROWS * DD / 4; i += NTHR) { dM[i] = srcM[i]; dX[i] = srcI[i]; }
    }
    __syncthreads();

    v8f nm[4];
    #pragma unroll
    for (int j = 0; j < 4; ++j) {
        const float bias = bl[nB + j * 16 + L];
        #pragma unroll
        for (int r = 0; r < 8; ++r) nm[j][r] = bias;
    }

    // K part 0: memory with Wl[:, 0:512]
    #pragma unroll 4
    for (int k0 = 0; k0 < DD; k0 += 4) {
        const v2f a = *(const v2f*)&sM[L * DD + k0 + 2 * hi];
        #pragma unroll
        for (int j = 0; j < 4; ++j) {
            const int n = nB + j * 16 + L;
            const v2f b = *(const v2f*)&Wl[(long)n * (3 * DD) + k0 + 2 * hi];
            nm[j] = wmma_f32(a, b, nm[j]);
        }
    }
    // K part 1: info with Wl[:, 512:1024]
    #pragma unroll 4
    for (int k0 = 0; k0 < DD; k0 += 4) {
        const v2f a = *(const v2f*)&sX[L * DD + k0 + 2 * hi];
        #pragma unroll
        for (int j = 0; j < 4; ++j) {
            const int n = nB + j * 16 + L;
            const v2f b = *(const v2f*)&Wl[(long)n * (3 * DD) + DD + k0 + 2 * hi];
            nm[j] = wmma_f32(a, b, nm[j]);
        }
    }
    __syncthreads();
    // stage selfSmry
    {
        const v4f* src = (const v4f*)(selfSmry + (long)rowBase * DD);
        v4f* dX = (v4f*)sX;
        #pragma unroll
        for (int i = tid; i < BROWS * DD / 4; i += NTHR) dX[i] = src[i];
    }
    __syncthreads();
    // K part 2: selfSmry with Wl[:, 1024:1536]
    #pragma unroll 4
    for (int k0 = 0; k0 < DD; k0 += 4) {
        const v2f a = *(const v2f*)&sX[L * DD + k0 + 2 * hi];
        #pragma unroll
        for (int j = 0; j < 4; ++j) {
            const int n = nB + j * 16 + L;
            const v2f b = *(const v2f*)&Wl[(long)n * (3 * DD) + 2 * DD + k0 + 2 * hi];
            nm[j] = wmma_f32(a, b, nm[j]);
        }
    }
    __syncthreads();
    // stage control
    {
        const v4f* src = (const v4f*)(control + (long)rowBase * DD);
        v4f* dX = (v4f*)sX;
        #pragma unroll
        for (int i = tid; i < BROWS * DD / 4; i += NTHR) dX[i] = src[i];
    }
    __syncthreads();

    // gate GEMM: z_pre = control @ Wg.T + bg
    v8f gz[4];
    #pragma unroll
    for (int j = 0; j < 4; ++j) {
        const float bias = bg[nB + j * 16 + L];
        #pragma unroll
        for (int r = 0; r < 8; ++r) gz[j][r] = bias;
    }
    #pragma unroll 4
    for (int k0 = 0; k0 < DD; k0 += 4) {
        const v2f a = *(const v2f*)&sX[L * DD + k0 + 2 * hi];
        #pragma unroll
        for (int j = 0; j < 4; ++j) {
            const int n = nB + j * 16 + L;
            const v2f b = *(const v2f*)&Wg[(long)n * DD + k0 + 2 * hi];
            gz[j] = wmma_f32(a, b, gz[j]);
        }
    }

    // epilogue: sigmoid gate + blend, write out
    #pragma unroll
    for (int j = 0; j < 4; ++j) {
        const int n = nB + j * 16 + L;
        #pragma unroll
        for (int r = 0; r < 8; ++r) {
            const int m = 8 * hi + r;
            const float z   = 1.f / (1.f + expf(-gz[j][r]));
            const float mem = sM[m * DD + n];
            out[(long)(rowBase + m) * DD + n] = nm[j][r] * z + mem * (1.f - z);
        }
    }
}

extern "C" void kernel_launch(void* const* d_in, const int* in_sizes, int n_in,
                              void* d_out, int out_size, void* d_ws, size_t ws_size,
                              hipStream_t stream) {
    const float* memory      = (const float*)d_in[0];
    const float* info        = (const float*)d_in[1];
    const float* control     = (const float*)d_in[2];
    const float* contControl = (const float*)d_in[3];
    const float* allControls = (const float*)d_in[4];
    const float* allMemories = (const float*)d_in[5];
    const float* Wc          = (const float*)d_in[6];
    const float* bc          = (const float*)d_in[7];
    const float* Wa          = (const float*)d_in[8];
    const float* ba          = (const float*)d_in[9];
    const float* Wl          = (const float*)d_in[10];
    const float* bl          = (const float*)d_in[11];
    const float* Wg          = (const float*)d_in[12];
    const float* bg          = (const float*)d_in[13];
    float* outp              = (float*)d_out;

    float* selfSmry = (float*)d_ws;            // B*D floats = 16 MB scratch

    const int nBlocks = BB / BROWS;            // 512

    wu_attn_kernel<<<nBlocks, NTHR, 0, stream>>>(
        contControl, allControls, allMemories, Wc, bc, Wa, ba, selfSmry);

    wu_write_kernel<<<nBlocks, NTHR, 0, stream>>>(
        memory, info, control, selfSmry, Wl, bl, Wg, bg, outp);
}